// NECTAR_Binning_48782238548322
// MI455X (gfx1250) — compile-verified
//
#include <hip/hip_runtime.h>
#include <hip/hip_bf16.h>

#define NUM_C   8
#define NBINS   15
#define TW      64                     // tile width (pixels)
#define TH      16                     // tile height
#define PX      4                      // pixels per thread (float4 along x)
#define BDX     (TW / PX)              // 16
#define BDY     TH                     // 16 -> 256 threads, 8 wave32
#define OP      (TW + 2)               // 66: one-hot LDS pitch
#define OROWS   (TH + 2)               // 18
#define N_HALO  (2 * OP + 2 * TH)      // 164 halo ring pixels
#define TABLE_N (NUM_C * 9 * NBINS)    // 1080

__global__ __launch_bounds__(256)
void NECTAR_Binning_48782238548322_kernel(const float* __restrict__ logits,
                                          const float* __restrict__ val_freqs,
                                          float* __restrict__ out,
                                          int H, int W)
{
    __shared__ float    s_vf[TABLE_N];          // calibration table (async-loaded)
    __shared__ unsigned s_o[OROWS * OP];        // nibble one-hot of hard label per pixel

    const int tx  = threadIdx.x;                // 0..15
    const int ty  = threadIdx.y;                // 0..15
    const int tid = ty * BDX + tx;              // 0..255
    const int x0  = blockIdx.x * TW + tx * PX;
    const int gy  = blockIdx.y * TH + ty;
    const int b   = blockIdx.z;
    const int HW  = H * W;

    const float* base = logits + (long)b * NUM_C * HW;

    // ---- gfx1250 async-to-LDS copy of the table (overlaps with softmax below)
    #pragma unroll
    for (int i = 0; i < 5; ++i) {
        int idx = i * 256 + tid;
        if (idx < TABLE_N) {
            unsigned lds_addr = (unsigned)(uintptr_t)(&s_vf[idx]);
            unsigned goff     = (unsigned)(idx * 4);
            asm volatile("global_load_async_to_lds_b32 %0, %1, %2 offset:0"
                         :
                         : "v"(lds_addr), "v"(goff), "s"(val_freqs)
                         : "memory");
        }
    }

    // ---- 8 classes x float4 (4 consecutive pixels) : b128 coalesced loads
    const int pix = gy * W + x0;
    float l[PX][NUM_C];
    #pragma unroll
    for (int c = 0; c < NUM_C; ++c) {
        float4 v = *(const float4*)(base + c * HW + pix);
        l[0][c] = v.x; l[1][c] = v.y; l[2][c] = v.z; l[3][c] = v.w;
    }

    // ---- per pixel: argmax (first-max tie-break), softmax, nibble-packed bins
    unsigned o_[PX];     // one-hot nibble of own hard label
    unsigned bp[PX];     // 8 bins packed 4b each
    #pragma unroll
    for (int i = 0; i < PX; ++i) {
        float m = l[i][0];
        int  am = 0;
        #pragma unroll
        for (int c = 1; c < NUM_C; ++c) {
            if (l[i][c] > m) { m = l[i][c]; am = c; }
        }
        float e[NUM_C];
        float s = 0.0f;
        #pragma unroll
        for (int c = 0; c < NUM_C; ++c) { e[c] = __expf(l[i][c] - m); s += e[c]; }
        float inv = 1.0f / s;                      // one IEEE div per pixel
        unsigned pack = 0;
        #pragma unroll
        for (int c = 0; c < NUM_C; ++c) {
            int bi = (int)(e[c] * inv * (float)NBINS);   // trunc == floor (p >= 0)
            bi = bi > (NBINS - 1) ? (NBINS - 1) : bi;
            pack |= (unsigned)bi << (4 * c);
        }
        bp[i] = pack;
        o_[i] = 1u << (4 * am);
        s_o[(ty + 1) * OP + (tx * PX + 1) + i] = o_[i];
    }

    // ---- halo ring: argmax only; outside image -> 0 one-hot (= SAME zero pad)
    if (tid < N_HALO) {
        int i, j;
        if      (tid < OP)           { i = tid;           j = 0;         }
        else if (tid < 2 * OP)       { i = tid - OP;      j = OROWS - 1; }
        else if (tid < 2 * OP + TH)  { i = 0;             j = tid - 2 * OP + 1; }
        else                         { i = OP - 1;        j = tid - (2 * OP + TH) + 1; }
        int hx = blockIdx.x * TW + i - 1;
        int hy = blockIdx.y * TH + j - 1;
        unsigned ov = 0;
        if (hx >= 0 && hx < W && hy >= 0 && hy < H) {
            int hp = hy * W + hx;
            float best = base[hp];
            int  lab = 0;
            #pragma unroll
            for (int c = 1; c < NUM_C; ++c) {
                float v = base[c * HW + hp];
                if (v > best) { best = v; lab = c; }
            }
            ov = 1u << (4 * lab);
        }
        s_o[j * OP + i] = ov;
    }

    asm volatile("s_wait_asynccnt 0" ::: "memory");
    __syncthreads();

    // ---- 3x3 one-hot box count, all 8 classes at once via nibble adds.
    // 6 overlapping values per row cover 4 pixels; max nibble value 9 < 16.
    unsigned cnt[PX] = {0u, 0u, 0u, 0u};
    #pragma unroll
    for (int dj = 0; dj < 3; ++dj) {
        unsigned q[PX + 2];
        #pragma unroll
        for (int k = 0; k < PX + 2; ++k)
            q[k] = s_o[(ty + dj) * OP + tx * PX + k];
        #pragma unroll
        for (int i = 0; i < PX; ++i)
            cnt[i] += q[i] + q[i + 1] + q[i + 2];
    }
    #pragma unroll
    for (int i = 0; i < PX; ++i) cnt[i] -= o_[i];   // exclude center -> 0..8 per nibble

    // ---- table gather, per-pixel normalization, b128 stores per class plane
    float f[NUM_C][PX];
    float tot[PX] = {0.f, 0.f, 0.f, 0.f};
    #pragma unroll
    for (int c = 0; c < NUM_C; ++c) {
        #pragma unroll
        for (int i = 0; i < PX; ++i) {
            int n  = (cnt[i] >> (4 * c)) & 0xF;     // 0..8
            int bi = (bp[i]  >> (4 * c)) & 0xF;     // 0..14
            float v = s_vf[c * (9 * NBINS) + n * NBINS + bi];
            f[c][i] = v;
            tot[i] += v;
        }
    }
    float invn[PX];
    #pragma unroll
    for (int i = 0; i < PX; ++i) {
        float t = (tot[i] == 0.0f) ? 1.0f : tot[i];
        invn[i] = 1.0f / t;
    }
    float* ob = out + (long)b * NUM_C * HW;
    #pragma unroll
    for (int c = 0; c < NUM_C; ++c) {
        float4 v;
        v.x = f[c][0] * invn[0];
        v.y = f[c][1] * invn[1];
        v.z = f[c][2] * invn[2];
        v.w = f[c][3] * invn[3];
        *(float4*)(ob + c * HW + pix) = v;
    }
}

extern "C" void kernel_launch(void* const* d_in, const int* in_sizes, int n_in,
                              void* d_out, int out_size, void* d_ws, size_t ws_size,
                              hipStream_t stream) {
    (void)n_in; (void)d_ws; (void)ws_size; (void)out_size; (void)in_sizes;
    const float* logits    = (const float*)d_in[0];   // [16, 8, 512, 512] f32
    const float* val_freqs = (const float*)d_in[1];   // [8, 9, 15] f32
    float*       out       = (float*)d_out;           // [16, 8, 512, 512] f32

    const int B = 16, H = 512, W = 512;
    dim3 block(BDX, BDY, 1);                 // 256 threads = 8 wave32
    dim3 grid(W / TW, H / TH, B);            // (8, 32, 16)
    NECTAR_Binning_48782238548322_kernel<<<grid, block, 0, stream>>>(logits, val_freqs, out, H, W);
}